// FFM_Layer_58780922413713
// MI455X (gfx1250) — compile-verified
//
#include <hip/hip_runtime.h>
#include <hip/hip_bf16.h>

// FFM forward, MI455X (gfx1250, wave32).
// Memory-bound gather-reduce: ~88.8 MB of random v-row gathers per call
// (~3.8 us floor at 23.3 TB/s). One wave32 per batch row; lane f streams the
// 832B v-row of feature f as 52 float4 loads; w[] scalars staged via
// global_load_async_to_lds_b32 (ASYNCcnt path); wave shuffle-tree combine.

#define FFM_F    26
#define FFM_K    8
#define FFM_ROWF (FFM_F * FFM_K)   // 208 floats per gathered v-row (832 B)
#define WAVES_PER_BLOCK 8

__global__ __launch_bounds__(WAVES_PER_BLOCK * 32)
void ffm_fwd_kernel(const int* __restrict__ inputs,    // [B, 26]
                    const int* __restrict__ offsets,   // [26]
                    const float* __restrict__ w0,      // [1]
                    const float* __restrict__ w,       // [520000, 1]
                    const float* __restrict__ v,       // [520000, 26, 8]
                    float* __restrict__ out,           // [B]
                    int batch)
{
    __shared__ float wstage[WAVES_PER_BLOCK][FFM_F];

    const int lane = threadIdx.x & 31;
    const int wave = threadIdx.x >> 5;
    const int b    = blockIdx.x * WAVES_PER_BLOCK + wave;
    if (b >= batch) return;

    const bool active = (lane < FFM_F);
    const int  f      = active ? lane : 0;
    const int  g      = inputs[b * FFM_F + f] + offsets[f];

    float a0 = 0.f, a1 = 0.f, a2 = 0.f, a3 = 0.f;
    float a4 = 0.f, a5 = 0.f, a6 = 0.f, a7 = 0.f;
    float psq = 0.f;

    if (active) {
        // Stage w[g] (scattered 4B gather) into LDS via the CDNA5 async-DMA
        // path; tracked by ASYNCcnt, consumed after the row-streaming loop.
        unsigned           lds_addr = (unsigned)(size_t)(&wstage[wave][lane]);
        unsigned long long gaddr    = (unsigned long long)(size_t)(w + g);
        asm volatile("global_load_async_to_lds_b32 %0, %1, off"
                     :: "v"(lds_addr), "v"(gaddr) : "memory");

        // Stream this lane's 832B v-row: 52 aligned float4 loads.
        // Row layout is [j=0..25][k=0..7] (k fastest), so even chunks hold
        // k=0..3 and odd chunks k=4..7: accumulators need no permutation.
        const float4* __restrict__ row =
            (const float4*)(v + (size_t)g * FFM_ROWF);
        __builtin_prefetch(row, 0, 1);   // global_prefetch_b8

        #pragma unroll
        for (int j = 0; j < FFM_F; ++j) {
            float4 lo = row[2 * j + 0];
            float4 hi = row[2 * j + 1];
            a0 += lo.x; a1 += lo.y; a2 += lo.z; a3 += lo.w;
            a4 += hi.x; a5 += hi.y; a6 += hi.z; a7 += hi.w;
        }
        // per-feature squared contribution: sum_k u_f[k]^2
        psq = a0*a0 + a1*a1 + a2*a2 + a3*a3 + a4*a4 + a5*a5 + a6*a6 + a7*a7;
    }

    // Wait for this wave's async LDS stage before reading wstage.
#if __has_builtin(__builtin_amdgcn_s_wait_asynccnt)
    __builtin_amdgcn_s_wait_asynccnt(0);
#else
    asm volatile("s_wait_asynccnt 0x0" ::: "memory");
#endif

    // Fold first-order term and squared term into one reduced scalar:
    //   out = w0 + sum_f(w_f) + 0.5*(S2 - sum_f psq_f)
    //       = w0 + sum_f(w_f - 0.5*psq_f) + 0.5*S2
    float t = active ? (wstage[wave][lane] - 0.5f * psq) : 0.f;

    // wave32 tree reduction of {u[0..7], t} (lanes >= 26 hold zeros).
    #pragma unroll
    for (int off = 16; off >= 1; off >>= 1) {
        a0 += __shfl_down(a0, off, 32);
        a1 += __shfl_down(a1, off, 32);
        a2 += __shfl_down(a2, off, 32);
        a3 += __shfl_down(a3, off, 32);
        a4 += __shfl_down(a4, off, 32);
        a5 += __shfl_down(a5, off, 32);
        a6 += __shfl_down(a6, off, 32);
        a7 += __shfl_down(a7, off, 32);
        t  += __shfl_down(t,  off, 32);
    }

    if (lane == 0) {
        // a0..a7 now hold s[k] = sum_f u_f[k]
        float s2 = a0*a0 + a1*a1 + a2*a2 + a3*a3 + a4*a4 + a5*a5 + a6*a6 + a7*a7;
        out[b] = w0[0] + t + 0.5f * s2;
    }
}

extern "C" void kernel_launch(void* const* d_in, const int* in_sizes, int n_in,
                              void* d_out, int out_size, void* d_ws, size_t ws_size,
                              hipStream_t stream) {
    const int*   inputs  = (const int*)d_in[0];    // [B, 26] int32
    const int*   offsets = (const int*)d_in[1];    // [26]    int32
    const float* w0      = (const float*)d_in[2];  // [1]     f32
    const float* w       = (const float*)d_in[3];  // [520000,1] f32
    const float* v       = (const float*)d_in[4];  // [520000,26,8] f32
    float*       out     = (float*)d_out;          // [B] f32

    const int B = in_sizes[0] / FFM_F;
    dim3 grid((B + WAVES_PER_BLOCK - 1) / WAVES_PER_BLOCK);
    dim3 block(WAVES_PER_BLOCK * 32);
    ffm_fwd_kernel<<<grid, block, 0, stream>>>(inputs, offsets, w0, w, v, out, B);
}